// Encoder_51453708206324
// MI455X (gfx1250) — compile-verified
//
#include <hip/hip_runtime.h>

#define T_WIN   16
#define NPOLE_  161
#define KDIM    644        // 4*NPOLE atoms
#define KPAD    672        // 42*16, zero-padded rows of A (21 M-tile pairs)
#define NP      2048       // pixels
#define MAXITER 100

typedef __attribute__((ext_vector_type(2))) float v2f;
typedef __attribute__((ext_vector_type(8))) float v8f;

// scalar slot layout in workspace:
// 0: lambd   1: t      2: tt      3: done
// 4: linv    5: done_prev (this iter)   6: diffsq accum   7: ||DtD||_F^2 accum

__global__ __launch_bounds__(32) void k_zero_scal(float* __restrict__ s) {
  if (threadIdx.x < 8) s[threadIdx.x] = 0.f;
}

// Dictionary: Dt[k][t] = normalized D^T  (atom-major, 16 contiguous floats per atom)
__global__ __launch_bounds__(256) void k_build_dic(const float* __restrict__ Drr,
                                                   const float* __restrict__ Dth,
                                                   float* __restrict__ Dt) {
  int k = blockIdx.x * blockDim.x + threadIdx.x;
  if (k >= KDIM) return;
  int g = k / NPOLE_;
  int p = k - g * NPOLE_;
  float r  = Drr[p];
  float th = Dth[p];
  float v[T_WIN];
  float pr = 1.f, sum = 0.f;
#pragma unroll
  for (int i = 0; i < T_WIN; ++i) {
    float fi = (float)i;
    float trig = (g >= 2) ? __sinf(fi * th) : __cosf(fi * th);
    float sg = ((g & 1) && (i & 1)) ? -1.f : 1.f;
    v[i] = sg * pr * trig;
    sum += v[i];
    pr *= r;
  }
  float mean = sum * (1.f / T_WIN);
  float ss = 0.f;
#pragma unroll
  for (int i = 0; i < T_WIN; ++i) { v[i] -= mean; ss += v[i] * v[i]; }
  float sd = __builtin_sqrtf(ss * (1.f / (T_WIN - 1)));   // ddof=1
  if (sd == 0.f) sd = 1.f;
  float inv = 1.f / sd;
#pragma unroll
  for (int i = 0; i < T_WIN; ++i) Dt[k * T_WIN + i] = v[i] * inv;
}

// DtD into A-buffer (first 644 rows) + Frobenius^2 via wave reduce + atomic
__global__ __launch_bounds__(256) void k_dtd(const float* __restrict__ Dt,
                                             float* __restrict__ A,
                                             float* __restrict__ s) {
  int idx = blockIdx.x * 256 + threadIdx.x;
  bool valid = idx < KDIM * KDIM;
  float acc = 0.f;
  if (valid) {
    int i = idx / KDIM, j = idx - i * KDIM;
    const float* a = Dt + i * T_WIN;
    const float* b = Dt + j * T_WIN;
#pragma unroll
    for (int t = 0; t < T_WIN; ++t) acc += a[t] * b[t];
    A[idx] = acc;
  }
  float v = acc * acc;
#pragma unroll
  for (int o = 16; o > 0; o >>= 1) v += __shfl_xor(v, o, 32);
  if ((threadIdx.x & 31) == 0) atomicAdd(&s[7], v);
}

__global__ void k_scal_init(float* __restrict__ s) {
  float linv = 1.f / __builtin_sqrtf(s[7]);
  s[0] = 0.1f * linv;   // lambd = LAM / L
  s[1] = 1.f;           // t
  s[2] = 0.f;           // tt
  s[3] = 0.f;           // done
  s[4] = linv;
  s[5] = 0.f;           // done_prev
  s[6] = 0.f;           // diffsq
}

// A = I - DtD*linv, rows [644,672) zeroed for padded WMMA tiles
__global__ __launch_bounds__(256) void k_a_final(float* __restrict__ A,
                                                 const float* __restrict__ s) {
  int idx = blockIdx.x * 256 + threadIdx.x;
  if (idx >= KPAD * KDIM) return;
  int i = idx / KDIM, j = idx - i * KDIM;
  float out = 0.f;
  if (i < KDIM) out = ((i == j) ? 1.f : 0.f) - A[idx] * s[4];
  A[idx] = out;
}

// DtY = linv * D^T x ; also init y = 0 and x_prev = 0
__global__ __launch_bounds__(256) void k_dty_init(const float* __restrict__ Dt,
                                                  const float* __restrict__ x,
                                                  const float* __restrict__ s,
                                                  float* __restrict__ DtY,
                                                  float* __restrict__ y,
                                                  float* __restrict__ xprev) {
  int idx = blockIdx.x * 256 + threadIdx.x;
  if (idx >= KDIM * NP) return;
  int k = idx / NP, p = idx - k * NP;
  const float* dk = Dt + k * T_WIN;
  float acc = 0.f;
#pragma unroll
  for (int t = 0; t < T_WIN; ++t) acc += dk[t] * x[t * NP + p];
  DtY[idx] = acc * s[4];
  y[idx] = 0.f;
  xprev[idx] = 0.f;
}

// One FISTA half-step: x_new = softshrink(A @ y + DtY, lambd) via f32 WMMA.
// Wave tile: 32(M) x 64(N) -> 8 accumulators; each B fragment feeds 2 WMMAs.
__global__ __launch_bounds__(256) void k_fista_gemm(const float* __restrict__ A,
                                                    const float* __restrict__ DtY,
                                                    const float* __restrict__ Y,
                                                    const float* __restrict__ xprev,
                                                    float* __restrict__ xcur,
                                                    float* __restrict__ s) {
  const int lane   = threadIdx.x & 31;
  const int waveId = blockIdx.x * (blockDim.x >> 5) + (threadIdx.x >> 5);
  const int mpair  = waveId >> 5;          // 0..20  (two 16-row M tiles each)
  const int ngrp   = waveId & 31;          // 0..31  (64 cols each)
  const int r16    = lane & 15;
  const int khalf  = lane >> 4;            // A-frag: VGPR0 = K{0,2}, VGPR1 = K{1,3}
  const int m0     = mpair * 32 + r16;     // < 672 (A padded)
  const int nb     = ngrp * 64 + r16;

  v8f acc[2][4] = {};
  const float* ap0 = A + (size_t)m0 * KDIM + 2 * khalf;
  const float* ap1 = ap0 + (size_t)16 * KDIM;
  for (int k0 = 0; k0 < KDIM; k0 += 4) {
    v2f a0 = *(const v2f*)(ap0 + k0);                        // M tile 0, K = 2*khalf..+1
    v2f a1 = *(const v2f*)(ap1 + k0);                        // M tile 1
    const float* yp = Y + (size_t)(k0 + 2 * khalf) * NP + nb;
#pragma unroll
    for (int j = 0; j < 4; ++j) {
      v2f b;
      b.x = yp[j * 16];                                      // B row K = 2*khalf
      b.y = yp[NP + j * 16];                                 // B row K = 2*khalf + 1
      acc[0][j] = __builtin_amdgcn_wmma_f32_16x16x4_f32(
          false, a0, false, b, (short)0, acc[0][j], false, false);
      acc[1][j] = __builtin_amdgcn_wmma_f32_16x16x4_f32(
          false, a1, false, b, (short)0, acc[1][j], false, false);
    }
  }

  const float lambd = s[0];
  float dsum = 0.f;
#pragma unroll
  for (int mt = 0; mt < 2; ++mt) {
#pragma unroll
    for (int r = 0; r < 8; ++r) {
      int mo = mpair * 32 + mt * 16 + khalf * 8 + r;         // C/D layout: M = r + 8*khalf
      if (mo < KDIM) {
        size_t base = (size_t)mo * NP + nb;
#pragma unroll
        for (int j = 0; j < 4; ++j) {
          float val = acc[mt][j][r] + DtY[base + j * 16];
          float mag = __builtin_fabsf(val) - lambd;
          float sh  = (mag > 0.f) ? ((val > 0.f) ? mag : -mag) : 0.f;
          float d   = sh - xprev[base + j * 16];
          dsum += d * d;
          xcur[base + j * 16] = sh;
        }
      }
    }
  }
#pragma unroll
  for (int o = 16; o > 0; o >>= 1) dsum += __shfl_xor(dsum, o, 32);
  if (lane == 0) atomicAdd(&s[6], dsum);
}

// Reference freeze semantics: done_prev gates this iteration's state update.
__global__ void k_scal_step(float* __restrict__ s) {
  float t    = s[1];
  float done = s[3];
  float diff = __builtin_sqrtf(s[6]) / (float)KDIM;
  float tnew = (1.f + __builtin_sqrtf(1.f + 4.f * t * t)) * 0.5f;
  float tt   = (t - 1.f) / tnew;
  bool frozen = (done != 0.f);
  s[5] = done;                                   // done_prev for elementwise kernel
  s[2] = frozen ? s[2] : tt;
  s[1] = frozen ? t : tnew;
  s[3] = (frozen || diff < 1e-4f) ? 1.f : 0.f;
  s[6] = 0.f;                                    // reset accumulator
}

// y_new = x_new*(1+tt) - x_old*tt  (frozen: keep y, restore x_cur = x_old)
__global__ __launch_bounds__(256) void k_update_y(float* __restrict__ xcur,
                                                  const float* __restrict__ xprev,
                                                  float* __restrict__ y,
                                                  const float* __restrict__ s) {
  int i = blockIdx.x * 256 + threadIdx.x;
  if (i >= KDIM * NP) return;
  if (s[5] != 0.f) {
    xcur[i] = xprev[i];
  } else {
    float tt = s[2];
    y[i] = xcur[i] * (1.f + tt) - xprev[i] * tt;
  }
}

extern "C" void kernel_launch(void* const* d_in, const int* in_sizes, int n_in,
                              void* d_out, int out_size, void* d_ws, size_t ws_size,
                              hipStream_t stream) {
  const float* Drr = (const float*)d_in[0];
  const float* Dth = (const float*)d_in[1];
  const float* x   = (const float*)d_in[2];
  float* out = (float*)d_out;
  float* ws  = (float*)d_ws;

  // workspace layout (floats, all 16-float aligned)
  float* Dt   = ws;                 // 644*16      = 10304
  float* Amat = ws + 10304;         // 672*644     = 432768
  float* DtY  = ws + 443072;        // 644*2048    = 1318912
  float* xb0  = ws + 1761984;       // 1318912
  float* xb1  = ws + 3080896;       // 1318912
  float* yb   = ws + 4399808;       // 1318912
  float* sc   = ws + 5718720;       // 8 scalars

  k_zero_scal<<<1, 32, 0, stream>>>(sc);
  k_build_dic<<<(KDIM + 255) / 256, 256, 0, stream>>>(Drr, Dth, Dt);
  k_dtd<<<(KDIM * KDIM + 255) / 256, 256, 0, stream>>>(Dt, Amat, sc);
  k_scal_init<<<1, 1, 0, stream>>>(sc);
  k_a_final<<<(KPAD * KDIM + 255) / 256, 256, 0, stream>>>(Amat, sc);
  k_dty_init<<<(KDIM * NP + 255) / 256, 256, 0, stream>>>(Dt, x, sc, DtY, yb, xb0);

  for (int it = 0; it < MAXITER; ++it) {
    const float* xp = (it & 1) ? xb1 : xb0;
    float* xc = (it == MAXITER - 1) ? out : ((it & 1) ? xb0 : xb1);
    // 21 M-pairs * 32 N-groups = 672 waves = 84 blocks * 8 waves
    k_fista_gemm<<<84, 256, 0, stream>>>(Amat, DtY, yb, xp, xc, sc);
    k_scal_step<<<1, 1, 0, stream>>>(sc);
    k_update_y<<<(KDIM * NP + 255) / 256, 256, 0, stream>>>(xc, xp, yb, sc);
  }
}